// GNNPathContext_71837622993133
// MI455X (gfx1250) — compile-verified
//
#include <hip/hip_runtime.h>
#include <hip/hip_bf16.h>

typedef float v2f __attribute__((ext_vector_type(2)));
typedef float v8f __attribute__((ext_vector_type(8)));

#define N_NODES 100000
#define N_EDGES 800000
#define NB      4000
#define GNN_IN  428
#define H       64
#define HID     512
#define EMB     300
#define CONCATD 1112   // HID + 2*EMB

// ---------------------------------------------------------------------------
// WMMA fp32 GEMM, 16x64 output strip per wave (4 N-tiles share one A fragment).
// C[M,N] = fuse(A[M,K](lda) @ B[K,N](ldb=N) + bias),  fuse: 0 none / 1 bias / 2 bias+relu
// Block = 128 threads = 4 waves, each wave a different 16-row M-tile.
// N must be a multiple of 64; n-block = blockIdx.y*64.
//
// V_WMMA_F32_16X16X4_F32 operand layout (ISA 7.12.2):
//   A 16x4 : lanes 0-15 hold {A[m0+l][k],A[m0+l][k+1]}, lanes 16-31 hold k+2,k+3
//   B 4x16 : lanes 0-15 hold {B[k][n+l],B[k+1][n+l]},   lanes 16-31 hold k+2,k+3
//   C/D    : vgpr v -> row m0+v (lanes 0-15) / m0+v+8 (lanes 16-31), col n+l
// ---------------------------------------------------------------------------
__global__ void gemm_wmma_f32(const float* __restrict__ A,
                              const float* __restrict__ B,
                              const float* __restrict__ bias,
                              float* __restrict__ C,
                              int M, int N, int K, int lda, int ldc, int fuse) {
    const int lane = threadIdx.x & 31;
    const int wave = threadIdx.x >> 5;
    const int m0 = (blockIdx.x * 4 + wave) * 16;
    if (m0 >= M) return;               // wave-uniform -> EXEC all-ones below
    const int n0 = blockIdx.y * 64;

    const int half = lane >> 4;        // 0: K pair {0,1}, 1: K pair {2,3}
    const int l    = lane & 15;

    // Induction-variable pointers: backend can use immediate-offset loads.
    const float* __restrict__ aptr = A + (size_t)(m0 + l) * lda + 2 * half;
    const float* __restrict__ bptr = B + (size_t)(2 * half) * N + n0 + l;
    const size_t bstep = (size_t)4 * N;

    v8f acc0 = {}, acc1 = {}, acc2 = {}, acc3 = {};

    for (int k = 0; k < K; k += 4) {
        v2f a;
        a.x = aptr[0];
        a.y = aptr[1];
        v2f b0, b1, b2, b3;
        b0.x = bptr[0];      b0.y = bptr[N];
        b1.x = bptr[16];     b1.y = bptr[N + 16];
        b2.x = bptr[32];     b2.y = bptr[N + 32];
        b3.x = bptr[48];     b3.y = bptr[N + 48];
        acc0 = __builtin_amdgcn_wmma_f32_16x16x4_f32(false, a, false, b0, (short)0, acc0, false, false);
        acc1 = __builtin_amdgcn_wmma_f32_16x16x4_f32(false, a, false, b1, (short)0, acc1, false, false);
        acc2 = __builtin_amdgcn_wmma_f32_16x16x4_f32(false, a, false, b2, (short)0, acc2, false, false);
        acc3 = __builtin_amdgcn_wmma_f32_16x16x4_f32(false, a, false, b3, (short)0, acc3, false, false);
        aptr += 4;
        bptr += bstep;
    }

    v8f accs[4] = {acc0, acc1, acc2, acc3};
#pragma unroll
    for (int nt = 0; nt < 4; ++nt) {
        const int col = n0 + nt * 16 + l;
        float bv = (fuse >= 1) ? bias[col] : 0.0f;
#pragma unroll
        for (int v = 0; v < 8; ++v) {
            int row = m0 + v + half * 8;
            float val = accs[nt][v] + bv;
            if (fuse == 2) val = fmaxf(val, 0.0f);
            C[(size_t)row * ldc + col] = val;
        }
    }
}

// ---------------------------------------------------------------------------
// Edge scatter in projected (64-dim) space: s[dst] += y[src].
// One wave per edge; lane handles features {lane, lane+32}.
// y (25.6 MB) fits in the 192 MB L2, so random gathers stay on-chip.
// ---------------------------------------------------------------------------
__global__ void edge_scatter64(const float* __restrict__ y,
                               float* __restrict__ s,
                               const int* __restrict__ src,
                               const int* __restrict__ dst, int nE) {
    int t = blockIdx.x * blockDim.x + threadIdx.x;
    int e = t >> 5;
    if (e >= nE) return;
    int lane = t & 31;
    size_t so  = (size_t)src[e] * H + lane;
    size_t dof = (size_t)dst[e] * H + lane;
    atomicAdd(&s[dof],      y[so]);
    atomicAdd(&s[dof + 32], y[so + 32]);
}

// z = relu(y + s + b1)   over [N_NODES, 64]
__global__ void add_bias_relu64(const float* __restrict__ y,
                                const float* __restrict__ s,
                                const float* __restrict__ b,
                                float* __restrict__ z, int total) {
    int t = blockIdx.x * blockDim.x + threadIdx.x;
    if (t >= total) return;
    z[t] = fmaxf(y[t] + s[t] + b[t & (H - 1)], 0.0f);
}

// pooled[batch[node]] += h[node];  counts[batch[node]] += 1
__global__ void pool_accum(const float* __restrict__ h,
                           const int* __restrict__ batch,
                           float* __restrict__ pooled,
                           float* __restrict__ counts, int total) {
    int t = blockIdx.x * blockDim.x + threadIdx.x;
    if (t >= total) return;
    int node = t >> 6;         // /H
    int f = t & (H - 1);
    int b = batch[node];
    atomicAdd(&pooled[(size_t)b * H + f], h[t]);
    if (f == 0) atomicAdd(&counts[b], 1.0f);
}

__global__ void pool_div(float* __restrict__ pooled,
                         const float* __restrict__ counts, int total) {
    int t = blockIdx.x * blockDim.x + threadIdx.x;
    if (t >= total) return;
    pooled[t] = pooled[t] / fmaxf(counts[t >> 6], 1.0f);
}

// concat[b, HID + j] = entity_embed[pred_pairs[b][j/EMB]][j%EMB], j in [0,600)
__global__ void gather_pairs(const int* __restrict__ pairs,
                             const float* __restrict__ emb,
                             float* __restrict__ cc, int total) {
    int t = blockIdx.x * blockDim.x + threadIdx.x;
    if (t >= total) return;
    int b = t / (2 * EMB);
    int j = t - b * (2 * EMB);
    int which = j / EMB;
    int k = j - which * EMB;
    int cls = pairs[b * 2 + which];
    cc[(size_t)b * CONCATD + HID + j] = emb[(size_t)cls * EMB + k];
}

// ---------------------------------------------------------------------------
extern "C" void kernel_launch(void* const* d_in, const int* in_sizes, int n_in,
                              void* d_out, int out_size, void* d_ws, size_t ws_size,
                              hipStream_t stream) {
    const int*   pred_pairs = (const int*)  d_in[0];
    const float* x          = (const float*)d_in[1];
    const int*   edge_index = (const int*)  d_in[2];
    const int*   batch_ids  = (const int*)  d_in[3];
    const float* ent_embed  = (const float*)d_in[4];
    const float* w1[3] = {(const float*)d_in[5],  (const float*)d_in[9],  (const float*)d_in[13]};
    const float* b1[3] = {(const float*)d_in[6],  (const float*)d_in[10], (const float*)d_in[14]};
    const float* w2[3] = {(const float*)d_in[7],  (const float*)d_in[11], (const float*)d_in[15]};
    const float* b2[3] = {(const float*)d_in[8],  (const float*)d_in[12], (const float*)d_in[16]};
    const float* lin_w   = (const float*)d_in[17];
    const float* lin_b   = (const float*)d_in[18];
    const float* final_w = (const float*)d_in[19];
    const float* final_b = (const float*)d_in[20];
    float* out = (float*)d_out;

    const int* e_src = edge_index;            // edge_index[0]
    const int* e_dst = edge_index + N_EDGES;  // edge_index[1]

    // Workspace layout (floats), ~122 MB total
    float* ws = (float*)d_ws;
    float* bufY   = ws;                         // N_NODES*H
    float* bufS   = bufY + (size_t)N_NODES * H; // N_NODES*H
    float* bufZ   = bufS + (size_t)N_NODES * H; // N_NODES*H
    float* bufH   = bufZ + (size_t)N_NODES * H; // N_NODES*H
    float* pooled = bufH + (size_t)N_NODES * H; // NB*H
    float* counts = pooled + (size_t)NB * H;    // NB
    float* cc     = counts + NB;                // NB*CONCATD

    const dim3 gblk(128);                        // 4 waves / block, 1 M-tile each
    const int  mblk_nodes = (N_NODES / 16 + 3) / 4;   // 1563 blocks over 6250 M-tiles
    const int  mblk_B     = (NB / 16 + 3) / 4;        // 63 blocks over 250 M-tiles
    const int  node_el    = N_NODES * H;              // 6.4M

    const float* hin = x;
    int kin = GNN_IN;
    for (int l = 0; l < 3; ++l) {
        // Y = Hin @ W1   (segment-sum moved after projection: matmul linearity,
        // so the edge scatter runs in 64-dim instead of 428-dim)
        gemm_wmma_f32<<<dim3(mblk_nodes, 1), gblk, 0, stream>>>(
            hin, w1[l], nullptr, bufY, N_NODES, H, kin, kin, H, /*fuse=*/0);
        // S = 0 ; S[dst] += Y[src]
        hipMemsetAsync(bufS, 0, (size_t)node_el * sizeof(float), stream);
        edge_scatter64<<<(N_EDGES * 32 + 255) / 256, 256, 0, stream>>>(
            bufY, bufS, e_src, e_dst, N_EDGES);
        // Z = relu(Y + S + b1)
        add_bias_relu64<<<(node_el + 255) / 256, 256, 0, stream>>>(
            bufY, bufS, b1[l], bufZ, node_el);
        // Hout = relu(Z @ W2 + b2)
        gemm_wmma_f32<<<dim3(mblk_nodes, 1), gblk, 0, stream>>>(
            bufZ, w2[l], b2[l], bufH, N_NODES, H, H, H, H, /*fuse=*/2);
        hin = bufH;
        kin = H;
    }

    // Segment-mean pooling
    hipMemsetAsync(pooled, 0, (size_t)(NB * H + NB) * sizeof(float), stream);
    pool_accum<<<(node_el + 255) / 256, 256, 0, stream>>>(
        bufH, batch_ids, pooled, counts, node_el);
    pool_div<<<(NB * H + 255) / 256, 256, 0, stream>>>(pooled, counts, NB * H);

    // emb = pooled @ lin_w + lin_b  -> concat columns [0, HID)
    gemm_wmma_f32<<<dim3(mblk_B, HID / 64), gblk, 0, stream>>>(
        pooled, lin_w, lin_b, cc, NB, HID, H, H, CONCATD, /*fuse=*/1);

    // concat columns [HID, HID+600): gathered class embeddings
    gather_pairs<<<(NB * 2 * EMB + 255) / 256, 256, 0, stream>>>(
        pred_pairs, ent_embed, cc, NB * 2 * EMB);

    // out = concat @ final_w + final_b
    gemm_wmma_f32<<<dim3(mblk_B, HID / 64), gblk, 0, stream>>>(
        cc, final_w, final_b, out, NB, HID, CONCATD, CONCATD, HID, /*fuse=*/1);
}